// GCNConvNet_88923002896921
// MI455X (gfx1250) — compile-verified
//
#include <hip/hip_runtime.h>

typedef float v2f __attribute__((ext_vector_type(2)));
typedef float v8f __attribute__((ext_vector_type(8)));

#define MAX_DEG 10
#define ALPHA   0.95f
#define BN_EPS  1e-5f

// ---------------------------------------------------------------- fill kernels
__global__ __launch_bounds__(256) void fill_f32(float* p, float v, int n) {
    int i = blockIdx.x * blockDim.x + threadIdx.x;
    if (i < n) p[i] = v;
}
__global__ __launch_bounds__(256) void fill_i32(int* p, int v, int n) {
    int i = blockIdx.x * blockDim.x + threadIdx.x;
    if (i < n) p[i] = v;
}

// ---------------------------------------------------------------- degree count
// Grid-stride + speculative prefetch of the next stride's indices
// (lowers to global_prefetch_b8; OOB prefetch is silently dropped).
__global__ __launch_bounds__(256) void count_deg(const long long* __restrict__ ei,
                                                 int* __restrict__ deg, int E) {
    const int stride = gridDim.x * blockDim.x;
    for (int e = blockIdx.x * blockDim.x + threadIdx.x; e < E; e += stride) {
        __builtin_prefetch(&ei[(long long)E + e + stride], 0, 1);
        int d = (int)ei[(long long)E + e];
        atomicAdd(&deg[d], 1);
    }
}

// ---------------------------------------------------------------- edge scatter-sum
// aggr[dst] += x[src], C channels per edge. aggr buffer is L2-resident (<20MB),
// so GLOBAL_ATOMIC_ADD_F32 resolves at L2 without HBM round trips. Edge-index
// streaming is the only real HBM traffic -> prefetch the next stride.
template <int C>
__global__ __launch_bounds__(256) void scatter_add(const long long* __restrict__ ei,
                                                   const float* __restrict__ x,
                                                   float* __restrict__ aggr, int E) {
    const int stride = gridDim.x * blockDim.x;
    for (int e = blockIdx.x * blockDim.x + threadIdx.x; e < E; e += stride) {
        __builtin_prefetch(&ei[e + stride], 0, 1);
        __builtin_prefetch(&ei[(long long)E + e + stride], 0, 1);
        int s = (int)ei[e];
        int d = (int)ei[(long long)E + e];
#pragma unroll
        for (int c = 0; c < C; ++c)
            atomicAdd(&aggr[d * C + c], x[s * C + c]);
    }
}

// ---------------------------------------------------------------- MFConv node transform
// h = alpha * (x @ Wl[deg] + bl[deg]) + (1-alpha) * (aggr @ Wr[deg]),
// then fused BN + ReLU. Per-degree weight tables cached in LDS when they fit
// under a safe 64KB static budget (CL: cache Wl, CR: cache Wr).
template <int CIN, int COUT, bool CL, bool CR>
__global__ __launch_bounds__(256) void node_transform(
    const float* __restrict__ x, const float* __restrict__ aggr,
    const int* __restrict__ deg,
    const float* __restrict__ Wl, const float* __restrict__ bl,
    const float* __restrict__ Wr,
    const float* __restrict__ gamma, const float* __restrict__ beta,
    const float* __restrict__ mean, const float* __restrict__ var,
    float* __restrict__ out, int n)
{
    constexpr int WSZ = (MAX_DEG + 1) * CIN * COUT;
    __shared__ float sWl[CL ? WSZ : 1];
    __shared__ float sWr[CR ? WSZ : 1];
    __shared__ float sbl[(MAX_DEG + 1) * COUT];

    for (int i = threadIdx.x; i < WSZ; i += blockDim.x) {
        if (CL) sWl[i] = Wl[i];
        if (CR) sWr[i] = Wr[i];
    }
    for (int i = threadIdx.x; i < (MAX_DEG + 1) * COUT; i += blockDim.x)
        sbl[i] = bl[i];
    __syncthreads();

    int node = blockIdx.x * blockDim.x + threadIdx.x;
    if (node >= n) return;

    int b = deg[node];
    b = b > MAX_DEG ? MAX_DEG : b;

    float xv[CIN], av[CIN];
#pragma unroll
    for (int i = 0; i < CIN; ++i) {
        xv[i] = x[node * CIN + i];
        av[i] = aggr[node * CIN + i];
    }

    const int wbase = b * CIN * COUT;
#pragma unroll
    for (int o = 0; o < COUT; ++o) {
        float s = sbl[b * COUT + o];
        float t = 0.f;
        const int base = wbase + o;
#pragma unroll
        for (int i = 0; i < CIN; ++i) {
            s += xv[i] * (CL ? sWl[base + i * COUT] : Wl[base + i * COUT]);
            t += av[i] * (CR ? sWr[base + i * COUT] : Wr[base + i * COUT]);
        }
        float h = ALPHA * s + (1.0f - ALPHA) * t;
        float z = (h - mean[o]) * rsqrtf(var[o] + BN_EPS) * gamma[o] + beta[o];
        out[node * COUT + o] = fmaxf(z, 0.f);   // ReLU before lin_s
    }
}

// ---------------------------------------------------------------- WMMA fp32 GEMM
// C[M,N] = act(A[M,K] @ W[K,N] + bias) via V_WMMA_F32_16X16X4_F32.
// K, N compile-time (12/24/48, 6..48): K-loop fully unrolls into a WMMA chain.
// W+bias staged once per block into LDS, zero-padded to 16-col tiles, so the
// inner loop is {global_load_b64 (A frag), 2x ds_load (B frag), v_wmma} with
// no EXEC masking. M must be a multiple of 16 (100000 = 6250*16); excess-wave
// exit is wave-uniform, so EXEC is all-ones at every WMMA (ISA requirement).
// ACT: 0 = none, 1 = relu, 2 = sigmoid.
template <int K, int N, int ACT>
__global__ __launch_bounds__(256) void gemm_wmma_f32(
    const float* __restrict__ A, const float* __restrict__ W,
    const float* __restrict__ bias, float* __restrict__ C, int M)
{
    constexpr int TILES_N = (N + 15) >> 4;
    constexpr int NPAD    = TILES_N << 4;
    constexpr bool FULL_N = (N % 16) == 0;

    __shared__ float sW[K * NPAD];
    __shared__ float sB[NPAD];

    for (int i = threadIdx.x; i < K * NPAD; i += blockDim.x) {
        int r = i / NPAD, c = i % NPAD;
        sW[i] = (c < N) ? W[r * N + c] : 0.f;
    }
    for (int i = threadIdx.x; i < NPAD; i += blockDim.x)
        sB[i] = (i < N) ? bias[i] : 0.f;
    __syncthreads();

    const int wave = blockIdx.x * (blockDim.x >> 5) + (threadIdx.x >> 5);
    const int tiles_m = M >> 4;                     // M is a multiple of 16
    if (wave >= tiles_m * TILES_N) return;          // wave-uniform exit

    const int tm = (wave / TILES_N) << 4;
    const int tn = (wave % TILES_N) << 4;
    const int lane = threadIdx.x & 31;
    const int half = lane >> 4;                     // 0: lanes 0-15, 1: 16-31
    const int l    = lane & 15;

    const int rowA = tm + l;                        // always < M
    const int colP = tn + l;                        // padded column, always valid

    v8f acc = {};
#pragma unroll
    for (int k = 0; k < K; k += 4) {
        const int ka = k + 2 * half;                // ISA 16x4 f32 A/B layout
        v2f a, b;
        a.x = A[rowA * K + ka];
        a.y = A[rowA * K + ka + 1];
        b.x = sW[ka * NPAD + colP];
        b.y = sW[(ka + 1) * NPAD + colP];
        acc = __builtin_amdgcn_wmma_f32_16x16x4_f32(
            false, a, false, b, (short)0, acc, false, false);
    }

    const float bv = sB[colP];
#pragma unroll
    for (int r = 0; r < 8; ++r) {
        const int m = tm + r + half * 8;            // ISA C/D f32 16x16 layout
        float v = acc[r] + bv;
        if (ACT == 1) v = fmaxf(v, 0.f);
        if (ACT == 2) v = 1.0f / (1.0f + __expf(-v));
        if (FULL_N || colP < N) C[m * N + colP] = v;
    }
}

// ---------------------------------------------------------------- launch
static inline int cdiv(int a, int b) { return (a + b - 1) / b; }

extern "C" void kernel_launch(void* const* d_in, const int* in_sizes, int n_in,
                              void* d_out, int out_size, void* d_ws, size_t ws_size,
                              hipStream_t stream) {
    const float*     x  = (const float*)d_in[0];
    const long long* ei = (const long long*)d_in[1];
    const int N = in_sizes[0] / 3;        // 100000
    const int E = in_sizes[1] / 2;        // 3200000

    // params, flattened in setup_inputs() insertion order
    const float* c1_Wl = (const float*)d_in[2];
    const float* c1_bl = (const float*)d_in[3];
    const float* c1_Wr = (const float*)d_in[4];
    const float* bn1_g = (const float*)d_in[5];
    const float* bn1_b = (const float*)d_in[6];
    const float* bn1_m = (const float*)d_in[7];
    const float* bn1_v = (const float*)d_in[8];
    const float* ls1_W = (const float*)d_in[9];
    const float* ls1_b = (const float*)d_in[10];
    const float* c2_Wl = (const float*)d_in[11];
    const float* c2_bl = (const float*)d_in[12];
    const float* c2_Wr = (const float*)d_in[13];
    const float* bn2_g = (const float*)d_in[14];
    const float* bn2_b = (const float*)d_in[15];
    const float* bn2_m = (const float*)d_in[16];
    const float* bn2_v = (const float*)d_in[17];
    const float* ls2_W = (const float*)d_in[18];
    const float* ls2_b = (const float*)d_in[19];
    const float* c3_Wl = (const float*)d_in[20];
    const float* c3_bl = (const float*)d_in[21];
    const float* c3_Wr = (const float*)d_in[22];
    const float* bn3_g = (const float*)d_in[23];
    const float* bn3_b = (const float*)d_in[24];
    const float* bn3_m = (const float*)d_in[25];
    const float* bn3_v = (const float*)d_in[26];
    const float* ls3_W = (const float*)d_in[27];
    const float* ls3_b = (const float*)d_in[28];
    const float* l1_W  = (const float*)d_in[29];
    const float* l1_b  = (const float*)d_in[30];
    const float* l2_W  = (const float*)d_in[31];
    const float* l2_b  = (const float*)d_in[32];
    const float* lo_W  = (const float*)d_in[33];
    const float* lo_b  = (const float*)d_in[34];

    // workspace carving: deg + three Nx48 fp32 buffers (~60MB total)
    char*  w   = (char*)d_ws;
    size_t off = 0;
    int*   deg = (int*)(w + off);
    off += ((size_t)N * sizeof(int) + 255) & ~(size_t)255;
    const size_t bufBytes = (((size_t)N * 48 * sizeof(float)) + 255) & ~(size_t)255;
    float* bufA = (float*)(w + off); off += bufBytes;
    float* bufB = (float*)(w + off); off += bufBytes;
    float* bufC = (float*)(w + off); off += bufBytes;

    const dim3 blk(256);
    const dim3 gE(cdiv(E, 256 * 4));      // grid-stride, 4 edges/thread
    const dim3 gN(cdiv(N, 256));

    // degrees (once, reused by all three conv layers)
    fill_i32<<<gN, blk, 0, stream>>>(deg, 0, N);
    count_deg<<<gE, blk, 0, stream>>>(ei, deg, E);

    auto gemmGrid = [&](int Ncol) {
        int tiles = (N >> 4) * cdiv(Ncol, 16);
        return dim3(cdiv(tiles, 8));      // 8 waves per 256-thread block
    };

    // ---- layer 1: cin=3 -> cout=12
    fill_f32<<<dim3(cdiv(N * 3, 256)), blk, 0, stream>>>(bufC, 0.f, N * 3);
    scatter_add<3><<<gE, blk, 0, stream>>>(ei, x, bufC, E);
    node_transform<3, 12, true, true><<<gN, blk, 0, stream>>>(
        x, bufC, deg, c1_Wl, c1_bl, c1_Wr, bn1_g, bn1_b, bn1_m, bn1_v, bufA, N);
    gemm_wmma_f32<12, 12, 1><<<gemmGrid(12), blk, 0, stream>>>(bufA, ls1_W, ls1_b, bufB, N);

    // ---- layer 2: cin=12 -> cout=24 (weights 25KB -> both in LDS)
    fill_f32<<<dim3(cdiv(N * 12, 256)), blk, 0, stream>>>(bufC, 0.f, N * 12);
    scatter_add<12><<<gE, blk, 0, stream>>>(ei, bufB, bufC, E);
    node_transform<12, 24, true, true><<<gN, blk, 0, stream>>>(
        bufB, bufC, deg, c2_Wl, c2_bl, c2_Wr, bn2_g, bn2_b, bn2_m, bn2_v, bufA, N);
    gemm_wmma_f32<24, 24, 1><<<gemmGrid(24), blk, 0, stream>>>(bufA, ls2_W, ls2_b, bufB, N);

    // ---- layer 3: cin=24 -> cout=48 (101KB total: Wr in LDS, Wl via L2)
    fill_f32<<<dim3(cdiv(N * 24, 256)), blk, 0, stream>>>(bufC, 0.f, N * 24);
    scatter_add<24><<<gE, blk, 0, stream>>>(ei, bufB, bufC, E);
    node_transform<24, 48, false, true><<<gN, blk, 0, stream>>>(
        bufB, bufC, deg, c3_Wl, c3_bl, c3_Wr, bn3_g, bn3_b, bn3_m, bn3_v, bufA, N);
    gemm_wmma_f32<48, 48, 1><<<gemmGrid(48), blk, 0, stream>>>(bufA, ls3_W, ls3_b, bufB, N);

    // ---- head MLP: 48 -> 24 -> 12 -> 6 (+sigmoid)
    gemm_wmma_f32<48, 24, 0><<<gemmGrid(24), blk, 0, stream>>>(bufB, l1_W, l1_b, bufA, N);
    gemm_wmma_f32<24, 12, 0><<<gemmGrid(12), blk, 0, stream>>>(bufA, l2_W, l2_b, bufC, N);
    gemm_wmma_f32<12, 6,  2><<<gemmGrid(6),  blk, 0, stream>>>(bufC, lo_W, lo_b, (float*)d_out, N);
}